// WaveletRNNCell_89017492177617
// MI455X (gfx1250) — compile-verified
//
#include <hip/hip_runtime.h>

typedef __attribute__((ext_vector_type(2))) float v2f;
typedef __attribute__((ext_vector_type(8))) float v8f;

#define TOT  512
#define HIDN 256
#define OUTN 128

// Compute 16 dot products (rows rowBase..rowBase+15 of W, each row TOT fp32,
// against s_inp[0..TOT)) using v_wmma_f32_16x16x4_f32 accumulation.
// A layout (ISA 7.12.2, 32-bit A 16x4): lane L, VGPR0/1 = W[row=L%16][base+2*(L/16)+{0,1}]
// B layout (symmetric with C/D row striping): lane L, VGPR0/1 = inp[base+2*(L/16)+{0,1}],
// identical across all 16 N columns -> every column of D holds the 16 row dots.
// Result: c[g] = dot(row rowBase+g) for lanes 0-15, dot(row rowBase+8+g) for lanes 16-31.
__device__ __forceinline__ v8f gemv16_wmma(const float* __restrict__ Wrow0,
                                           const float* s_inp, int lane) {
    const int half = lane >> 4;          // 0: K={0,1}+..., 1: K={2,3}+...
    const int m    = lane & 15;          // A-matrix row within tile
    const float* a_ptr = Wrow0 + (size_t)m * TOT + 2 * half;
    const float* b_ptr = s_inp + 2 * half;
    v8f c = {};
    // unroll 16 -> clause of 16 global_load_b64 in flight per wave (4 KB/wave)
    // to cover HBM latency on the read-once weight stream.
#pragma unroll 16
    for (int base = 0; base < TOT; base += 4) {
        v2f a = *(const v2f*)(a_ptr + base);   // global_load_b64, 8B aligned
        v2f b = *(const v2f*)(b_ptr + base);   // ds_load_b64,     8B aligned
        c = __builtin_amdgcn_wmma_f32_16x16x4_f32(
                /*neg_a=*/false, a, /*neg_b=*/false, b,
                /*c_mod=*/(short)0, c, /*reuse_a=*/false, /*reuse_b=*/false);
    }
    return c;
}

__global__ void __launch_bounds__(256)
wavelet_cell_kernel(const float* __restrict__ h,  const float* __restrict__ y,
                    const float* __restrict__ u,  const float* __restrict__ W1,
                    const float* __restrict__ b1, const float* __restrict__ W2,
                    const float* __restrict__ b2, const float* __restrict__ Wy,
                    const float* __restrict__ by, float* __restrict__ out) {
    __shared__ float s_inp[TOT];
    __shared__ float s_red[8];

    const int tid = threadIdx.x;
    // Stage inp = concat(h, y, u) into LDS (2 elements per thread).
    for (int i = tid; i < TOT; i += 256) {
        float v;
        if (i < HIDN)             v = h[i];
        else if (i < HIDN + OUTN) v = y[i - HIDN];
        else                      v = u[i - HIDN - OUTN];
        s_inp[i] = v;
    }
    __syncthreads();

    const int lane = tid & 31;
    const int wave = tid >> 5;
    const int half = lane >> 4;
    const int bid  = blockIdx.x;

    if (bid < HIDN - 1) {
        // ---- a2[bid]: product over 512 Mexican-hat activations of W2[bid] @ inp + b2[bid]
        const float* Wk = W2 + (size_t)bid * TOT * TOT;
        const float* bk = b2 + (size_t)bid * TOT;
        float wp = 1.0f;
#pragma unroll
        for (int grp = 0; grp < 4; ++grp) {
            const int rowBase = wave * 64 + grp * 16;
            v8f c = gemv16_wmma(Wk + (size_t)rowBase * TOT, s_inp, lane);
            float p = 1.0f;
#pragma unroll
            for (int g = 0; g < 8; ++g) {
                const float hh = c[g] + bk[rowBase + 8 * half + g];
                const float x2 = hh * hh;
                p *= (1.0f - x2) * __expf(-0.5f * x2);
            }
            p *= __shfl_xor(p, 16, 32);   // combine the two half-wave row sets
            wp *= p;
        }
        if (lane == 0) s_red[wave] = wp;
        __syncthreads();
        if (tid == 0) {
            float r = 1.0f;
#pragma unroll
            for (int w = 0; w < 8; ++w) r *= s_red[w];
            out[1 + bid] = r;
        }
    } else if (bid == HIDN - 1) {
        // ---- a1 = prod exp(-0.5*h11^2) = exp(-0.5 * sum h11^2), h11 = W1 @ inp + b1
        float ws = 0.0f;
#pragma unroll
        for (int grp = 0; grp < 4; ++grp) {
            const int rowBase = wave * 64 + grp * 16;
            v8f c = gemv16_wmma(W1 + (size_t)rowBase * TOT, s_inp, lane);
            float s = 0.0f;
#pragma unroll
            for (int g = 0; g < 8; ++g) {
                const float hh = c[g] + b1[rowBase + 8 * half + g];
                s += hh * hh;
            }
            s += __shfl_xor(s, 16, 32);
            ws += s;
        }
        if (lane == 0) s_red[wave] = ws;
        __syncthreads();
        if (tid == 0) {
            float s = 0.0f;
#pragma unroll
            for (int w = 0; w < 8; ++w) s += s_red[w];
            out[0] = __expf(-0.5f * s);
        }
    } else {
        // ---- y_out = Wy @ inp + by : 128 rows, one 16-row tile per wave
        const int rowBase = wave * 16;
        v8f c = gemv16_wmma(Wy + (size_t)rowBase * TOT, s_inp, lane);
        if ((lane & 15) == 0) {            // lanes 0 and 16 hold the results
#pragma unroll
            for (int g = 0; g < 8; ++g) {
                const int r = rowBase + 8 * half + g;
                out[HIDN + r] = c[g] + by[r];
            }
        }
    }
}

extern "C" void kernel_launch(void* const* d_in, const int* in_sizes, int n_in,
                              void* d_out, int out_size, void* d_ws, size_t ws_size,
                              hipStream_t stream) {
    (void)in_sizes; (void)n_in; (void)d_ws; (void)ws_size; (void)out_size;
    const float* h  = (const float*)d_in[0];
    const float* y  = (const float*)d_in[1];
    const float* u  = (const float*)d_in[2];
    const float* W1 = (const float*)d_in[3];
    const float* b1 = (const float*)d_in[4];
    const float* W2 = (const float*)d_in[5];
    const float* b2 = (const float*)d_in[6];
    const float* Wy = (const float*)d_in[7];
    const float* by = (const float*)d_in[8];

    dim3 grid(HIDN + 1);   // 255 x a2, 1 x a1, 1 x y_out
    dim3 block(256);       // 8 wave32 per WG
    hipLaunchKernelGGL(wavelet_cell_kernel, grid, block, 0, stream,
                       h, y, u, W1, b1, W2, b2, Wy, by, (float*)d_out);
}